// ConceptNet_New_43636867728061
// MI455X (gfx1250) — compile-verified
//
#include <hip/hip_runtime.h>

typedef __attribute__((ext_vector_type(2))) float v2f;
typedef __attribute__((ext_vector_type(8))) float v8f;

#define D_DIM 768
#define C_DIM 64
#define N_ROWS 50000
#define BS_ROWS 1024
#define NCLS 10
#define KNN 10
#define FLT_BIG 3.402823466e38f

// workspace layout (float offsets)
#define WS_DOT    0
#define WS_SQ     (N_ROWS * C_DIM)                 // 3,200,000
#define WS_GRAM   (WS_SQ + N_ROWS)                 // +50,000
#define WS_GINV   (WS_GRAM + C_DIM * C_DIM)
#define WS_T1     (WS_GINV + C_DIM * C_DIM)
#define WS_T2     (WS_T1 + BS_ROWS * C_DIM)
#define WS_CW     (WS_T2 + BS_ROWS * C_DIM)
#define WS_COLSUM (WS_CW + C_DIM * NCLS)
#define WS_BPACK  (WS_COLSUM + C_DIM)              // 768*64 floats, K-pair packed

// ---------------------------------------------------------------------------
// Pack concept [768 x 64] row-major into K-pair-interleaved float2 layout:
// bpack[kp*64 + n] = (C[2kp][n], C[2kp+1][n]) -> one b64 per B fragment.
// ---------------------------------------------------------------------------
__global__ __launch_bounds__(256) void pack_b_kernel(const float* __restrict__ Cmat,
                                                     float* __restrict__ bpack) {
  const int idx = blockIdx.x * 256 + threadIdx.x;   // over 384*64 = 24576
  if (idx >= (D_DIM / 2) * C_DIM) return;
  const int kp = idx >> 6, n = idx & 63;
  float2 v;
  v.x = Cmat[(2 * kp) * C_DIM + n];
  v.y = Cmat[(2 * kp + 1) * C_DIM + n];
  ((float2*)bpack)[idx] = v;
}

// ---------------------------------------------------------------------------
// GEMM: rows[16 per block] x concept -> dot tile, V_WMMA_F32_16X16X4_F32.
// A strip (16 x 768) staged to LDS once; 4 waves, one 16-col tile each.
// Packed-B (b64 fragments) + dual accumulators to break the D->C WMMA chain.
// Optionally emits per-row squared norms (for the distance ranking key).
// ---------------------------------------------------------------------------
__global__ __launch_bounds__(128) void gemm_rows_x_concept(
    const float* __restrict__ E, const float* __restrict__ bpack,
    float* __restrict__ dot_out, float* __restrict__ sq_out) {
  __shared__ float4 lds4[16 * (D_DIM / 4)];     // 48 KB
  __shared__ float part[16][8];
  float* lds = (float*)lds4;

  const int tid  = threadIdx.x;
  const int row0 = blockIdx.x * 16;

  // stage A strip, coalesced 16B loads
  const float4* src = (const float4*)(E + (size_t)row0 * D_DIM);
  for (int i = tid; i < 16 * (D_DIM / 4); i += 128) lds4[i] = src[i];
  __syncthreads();

  if (sq_out) {
    const int r = tid >> 3, j = tid & 7;
    float s = 0.f;
    for (int k = j; k < D_DIM; k += 8) { float v = lds[r * D_DIM + k]; s += v * v; }
    part[r][j] = s;
    __syncthreads();
    if (tid < 16) {
      float t = 0.f;
      #pragma unroll
      for (int j2 = 0; j2 < 8; ++j2) t += part[tid][j2];
      sq_out[row0 + tid] = t;
    }
  }

  const int lane = tid & 31;
  const int wave = tid >> 5;
  const int n0 = wave * 16;
  const int r  = lane & 15;     // A row within tile
  const int p  = lane >> 4;     // K-pair selector (K = 2p, 2p+1)
  const int n  = lane & 15;     // B col within tile

  v8f acc0 = {}, acc1 = {};
  const float* arow = lds + r * D_DIM + 2 * p;
  const v2f* bp = (const v2f*)bpack + p * C_DIM + n0 + n;

  #pragma unroll 2
  for (int kb = 0; kb < D_DIM; kb += 8) {
    v2f a0, a1;
    a0.x = arow[kb];     a0.y = arow[kb + 1];
    a1.x = arow[kb + 4]; a1.y = arow[kb + 5];
    v2f b0 = bp[(kb >> 1) * C_DIM];                 // K-pair kb/2 + p
    v2f b1 = bp[(kb >> 1) * C_DIM + 2 * C_DIM];     // K-pair kb/2 + 2 + p
    acc0 = __builtin_amdgcn_wmma_f32_16x16x4_f32(false, a0, false, b0,
                                                 (short)0, acc0, false, false);
    acc1 = __builtin_amdgcn_wmma_f32_16x16x4_f32(false, a1, false, b1,
                                                 (short)0, acc1, false, false);
  }

  // D layout: VGPR v -> row (v + 8p), col n within tile
  float* drow = dot_out + (size_t)(row0 + 8 * p) * C_DIM + n0 + n;
  #pragma unroll
  for (int v = 0; v < 8; ++v) drow[(size_t)v * C_DIM] = acc0[v] + acc1[v];
}

// ---------------------------------------------------------------------------
// gram = C^T C  [64x64], WMMA, one wave per 16x16 tile (16 blocks)
// ---------------------------------------------------------------------------
__global__ __launch_bounds__(32) void gram_wmma(const float* __restrict__ Cmat,
                                                float* __restrict__ gram) {
  const int lane = threadIdx.x;
  const int m0 = (blockIdx.x >> 2) * 16;
  const int n0 = (blockIdx.x & 3) * 16;
  const int r = lane & 15, p = lane >> 4, n = lane & 15;

  v8f acc = {};
  const float* ap = Cmat + (2 * p) * C_DIM + m0 + r;  // A[m][k] = C[k][m]
  const float* bp = Cmat + (2 * p) * C_DIM + n0 + n;

  #pragma unroll 4
  for (int kb = 0; kb < D_DIM; kb += 4) {
    v2f a, b;
    a.x = ap[kb * C_DIM];
    a.y = ap[(kb + 1) * C_DIM];
    b.x = bp[kb * C_DIM];
    b.y = bp[(kb + 1) * C_DIM];
    acc = __builtin_amdgcn_wmma_f32_16x16x4_f32(false, a, false, b,
                                                (short)0, acc, false, false);
  }
  float* g = gram + (m0 + 8 * p) * C_DIM + n0 + n;
  #pragma unroll
  for (int v = 0; v < 8; ++v) g[v * C_DIM] = acc[v];
}

// L2 = (sum - trace)/4096, norm = trace/4096
__global__ __launch_bounds__(256) void gram_stats(const float* __restrict__ gram,
                                                  float* __restrict__ out) {
  __shared__ float ssum[256], strc[256];
  const int tid = threadIdx.x;
  float s = 0.f, t = 0.f;
  for (int i = tid; i < C_DIM * C_DIM; i += 256) {
    float g = gram[i];
    s += g;
    if ((i >> 6) == (i & 63)) t += g;
  }
  ssum[tid] = s; strc[tid] = t;
  __syncthreads();
  for (int st = 128; st > 0; st >>= 1) {
    if (tid < st) { ssum[tid] += ssum[tid + st]; strc[tid] += strc[tid + st]; }
    __syncthreads();
  }
  if (tid == 0) {
    out[BS_ROWS * NCLS + 1] = (ssum[0] - strc[0]) / 4096.0f;
    out[BS_ROWS * NCLS + 2] = strc[0] / 4096.0f;
  }
}

// ---------------------------------------------------------------------------
// 64x64 Gauss-Jordan inverse (SPD gram -> no pivoting), single block in LDS
// ---------------------------------------------------------------------------
__global__ __launch_bounds__(256) void inv_kernel(const float* __restrict__ gram,
                                                  float* __restrict__ ginv) {
  __shared__ float a[64][129];   // 64 x 128 augmented, +1 pad
  __shared__ float fac[64];
  const int tid = threadIdx.x;

  for (int idx = tid; idx < 64 * 128; idx += 256) {
    int r = idx >> 7, c = idx & 127;
    a[r][c] = (c < 64) ? gram[r * 64 + c] : ((c - 64) == r ? 1.0f : 0.0f);
  }
  __syncthreads();

  for (int k = 0; k < 64; ++k) {
    const float piv = a[k][k];
    __syncthreads();
    const float pinv = 1.0f / piv;
    if (tid < 128) a[k][tid] *= pinv;
    __syncthreads();
    if (tid < 64) fac[tid] = a[tid][k];
    __syncthreads();
    for (int idx = tid; idx < 64 * 128; idx += 256) {
      int r = idx >> 7, c = idx & 127;
      if (r != k) a[r][c] -= fac[r] * a[k][c];
    }
    __syncthreads();
  }

  for (int idx = tid; idx < 64 * 64; idx += 256) {
    int r = idx >> 6, c = idx & 63;
    ginv[r * 64 + c] = a[r][64 + c];
  }
}

// ---------------------------------------------------------------------------
// top-10 smallest d2 per concept column; key = ||e||^2 - 2*dot (col const drops).
// Sum the dot values of the winners -> colsum[c]. Deterministic (no atomics).
// ---------------------------------------------------------------------------
__global__ __launch_bounds__(256) void topk_kernel(const float* __restrict__ dot,
                                                   const float* __restrict__ sq,
                                                   float* __restrict__ colsum) {
  const int c = blockIdx.x;
  const int tid = threadIdx.x;

  float key[KNN], val[KNN];
  #pragma unroll
  for (int i = 0; i < KNN; ++i) { key[i] = FLT_BIG; val[i] = 0.f; }

  for (int n = tid; n < N_ROWS; n += 256) {
    const float d = dot[(size_t)n * C_DIM + c];
    const float k = sq[n] - 2.0f * d;
    if (k < key[KNN - 1]) {
      key[KNN - 1] = k; val[KNN - 1] = d;
      #pragma unroll
      for (int i = KNN - 1; i > 0; --i) {
        if (key[i] < key[i - 1]) {
          float tk = key[i]; key[i] = key[i - 1]; key[i - 1] = tk;
          float tv = val[i]; val[i] = val[i - 1]; val[i - 1] = tv;
        }
      }
    }
  }

  __shared__ float sk[256 * KNN];
  __shared__ float sv[256 * KNN];
  __shared__ float rk[256];
  __shared__ int   ri[256];
  #pragma unroll
  for (int i = 0; i < KNN; ++i) { sk[tid * KNN + i] = key[i]; sv[tid * KNN + i] = val[i]; }
  __syncthreads();

  __shared__ float csum_s;
  if (tid == 0) csum_s = 0.f;
  __syncthreads();

  for (int round = 0; round < KNN; ++round) {
    float mk = FLT_BIG; int mi = 0;
    #pragma unroll
    for (int i = 0; i < KNN; ++i) {
      int idx = tid * KNN + i;
      if (sk[idx] < mk) { mk = sk[idx]; mi = idx; }
    }
    rk[tid] = mk; ri[tid] = mi;
    __syncthreads();
    for (int st = 128; st > 0; st >>= 1) {
      if (tid < st && rk[tid + st] < rk[tid]) { rk[tid] = rk[tid + st]; ri[tid] = ri[tid + st]; }
      __syncthreads();
    }
    if (tid == 0) { csum_s += sv[ri[0]]; sk[ri[0]] = FLT_BIG; }
    __syncthreads();
  }
  if (tid == 0) colsum[c] = csum_s;
}

__global__ __launch_bounds__(64) void l1_reduce(const float* __restrict__ colsum,
                                                float* __restrict__ out_l1) {
  __shared__ float s[64];
  const int t = threadIdx.x;
  s[t] = colsum[t];
  __syncthreads();
  for (int st = 32; st > 0; st >>= 1) {
    if (t < st) s[t] += s[t + st];
    __syncthreads();
  }
  if (t == 0) *out_l1 = s[0] / (float)(C_DIM * KNN);
}

// T2 = T1 @ Ginv   [1024 x 64], K = 64
__global__ __launch_bounds__(64) void t2_kernel(const float* __restrict__ t1,
                                                const float* __restrict__ ginv,
                                                float* __restrict__ t2o) {
  __shared__ float rowv[64];
  const int row = blockIdx.x, j = threadIdx.x;
  rowv[j] = t1[row * 64 + j];
  __syncthreads();
  float s = 0.f;
  #pragma unroll 8
  for (int k = 0; k < 64; ++k) s += rowv[k] * ginv[k * 64 + j];
  t2o[row * 64 + j] = s;
}

// CW = C^T @ W   [64 x 10], K = 768
__global__ __launch_bounds__(640) void cw_kernel(const float* __restrict__ Cmat,
                                                 const float* __restrict__ W,
                                                 float* __restrict__ cw) {
  const int tid = threadIdx.x;
  if (tid >= C_DIM * NCLS) return;
  const int c = tid / NCLS, o = tid % NCLS;
  float s = 0.f;
  for (int d = 0; d < D_DIM; ++d) s += Cmat[d * C_DIM + c] * W[d * NCLS + o];
  cw[c * NCLS + o] = s;
}

// y_pred = T2 @ CW + b   [1024 x 10], K = 64
__global__ __launch_bounds__(256) void ypred_kernel(const float* __restrict__ t2v,
                                                    const float* __restrict__ cw,
                                                    const float* __restrict__ b,
                                                    float* __restrict__ out) {
  const int idx = blockIdx.x * blockDim.x + threadIdx.x;
  if (idx >= BS_ROWS * NCLS) return;
  const int row = idx / NCLS, o = idx % NCLS;
  float s = b[o];
  #pragma unroll 8
  for (int c = 0; c < C_DIM; ++c) s += t2v[row * C_DIM + c] * cw[c * NCLS + o];
  out[idx] = s;
}

extern "C" void kernel_launch(void* const* d_in, const int* in_sizes, int n_in,
                              void* d_out, int out_size, void* d_ws, size_t ws_size,
                              hipStream_t stream) {
  const float* X  = (const float*)d_in[0];   // train_embedding  [1024,768]
  const float* E  = (const float*)d_in[1];   // train_embeddings [50000,768]
  const float* Cm = (const float*)d_in[2];   // concept          [768,64]
  const float* W  = (const float*)d_in[3];   // W_hx             [768,10]
  const float* b  = (const float*)d_in[4];   // b_hx             [10]
  float* out = (float*)d_out;
  float* ws  = (float*)d_ws;

  float* dot    = ws + WS_DOT;
  float* sq     = ws + WS_SQ;
  float* gram   = ws + WS_GRAM;
  float* ginv   = ws + WS_GINV;
  float* t1     = ws + WS_T1;
  float* t2     = ws + WS_T2;
  float* cw     = ws + WS_CW;
  float* colsum = ws + WS_COLSUM;
  float* bpack  = ws + WS_BPACK;

  pack_b_kernel<<<96, 256, 0, stream>>>(Cm, bpack);
  gram_wmma<<<16, 32, 0, stream>>>(Cm, gram);
  gram_stats<<<1, 256, 0, stream>>>(gram, out);
  inv_kernel<<<1, 256, 0, stream>>>(gram, ginv);

  gemm_rows_x_concept<<<N_ROWS / 16, 128, 0, stream>>>(E, bpack, dot, sq);
  gemm_rows_x_concept<<<BS_ROWS / 16, 128, 0, stream>>>(X, bpack, t1, nullptr);

  topk_kernel<<<C_DIM, 256, 0, stream>>>(dot, sq, colsum);
  l1_reduce<<<1, 64, 0, stream>>>(colsum, out + BS_ROWS * NCLS);

  t2_kernel<<<BS_ROWS, 64, 0, stream>>>(t1, ginv, t2);
  cw_kernel<<<1, 640, 0, stream>>>(Cm, W, cw);
  ypred_kernel<<<40, 256, 0, stream>>>(t2, cw, b, out);
}